// EHTMLoss_33732673143370
// MI455X (gfx1250) — compile-verified
//
#include <hip/hip_runtime.h>
#include <math.h>

#define B_SZ 1024
#define D_SZ 2048
#define N_SZ 3072            // 3*B
#define NC   64
#define MARGIN_F 0.8f
#define BIG_F 1e30f

typedef float v2f __attribute__((ext_vector_type(2)));
typedef float v8f __attribute__((ext_vector_type(8)));

// ---------------- init: zero accumulator region ----------------
__global__ void ehtm_init(float* __restrict__ p, int n) {
    int i = blockIdx.x * blockDim.x + threadIdx.x;
    if (i < n) p[i] = 0.0f;
}

// ---------------- pack emb = concat(anchor, positive, negative) ----------------
__global__ void ehtm_pack(const float* __restrict__ a, const float* __restrict__ p,
                          const float* __restrict__ n, float* __restrict__ emb) {
    int r = blockIdx.y;
    int c = blockIdx.x * blockDim.x + threadIdx.x;
    const float* src;
    if (r < B_SZ)            src = a + (size_t)r * D_SZ;
    else if (r < 2 * B_SZ)   src = p + (size_t)(r - B_SZ) * D_SZ;
    else                     src = n + (size_t)(r - 2 * B_SZ) * D_SZ;
    emb[(size_t)r * D_SZ + c] = src[c];
}

// ---------------- labels / class counts / in_anchor ----------------
__global__ void ehtm_labels(const int* __restrict__ al, const int* __restrict__ nl,
                            int* __restrict__ labels, float* __restrict__ counts,
                            int* __restrict__ in_anchor) {
    int r = blockIdx.x * blockDim.x + threadIdx.x;
    if (r >= N_SZ) return;
    int l = (r < 2 * B_SZ) ? al[r & (B_SZ - 1)] : nl[r - 2 * B_SZ];
    labels[r] = l;
    atomicAdd(&counts[l], 1.0f);
    if (r < B_SZ) in_anchor[l] = 1;
}

// ---------------- row norms + per-row triplet term ----------------
__global__ void ehtm_norms(const float* __restrict__ emb, float* __restrict__ sq,
                           float* __restrict__ scal) {
    __shared__ float red[3 * 256];
    int r = blockIdx.x, t = threadIdx.x;
    const float* row = emb + (size_t)r * D_SZ;
    bool isA = (r < B_SZ);
    const float* prow = emb + (size_t)(r + B_SZ) * D_SZ;
    const float* nrow = emb + (size_t)(r + 2 * B_SZ) * D_SZ;
    float s = 0.f, pd = 0.f, nd = 0.f;
    for (int c = t; c < D_SZ; c += 256) {
        float x = row[c];
        s = fmaf(x, x, s);
        if (isA) {
            float dp = x - prow[c]; pd = fmaf(dp, dp, pd);
            float dn = x - nrow[c]; nd = fmaf(dn, dn, nd);
        }
    }
    red[t] = s; red[256 + t] = pd; red[512 + t] = nd;
    __syncthreads();
    for (int o = 128; o > 0; o >>= 1) {
        if (t < o) {
            red[t] += red[t + o];
            red[256 + t] += red[256 + t + o];
            red[512 + t] += red[512 + t + o];
        }
        __syncthreads();
    }
    if (t == 0) {
        sq[r] = red[0];
        if (isA) {
            float tl = red[256] - red[512] + MARGIN_F;
            atomicAdd(&scal[0], fmaxf(tl, 0.f));
        }
    }
}

// ---------------- WMMA fp32 GEMM (G = E*E^T), symmetric, 32x32 per wave ----------------
// One wave computes a 32x32 output block (upper triangle only; mirrored stores).
// 4 accumulators reuse each A/B load across 2 WMMAs -> 1 b64 load per WMMA.
// A 16x4 f32 layout: lanes 0-15 hold {K=0,K=1}, lanes 16-31 hold {K=2,K=3}, M=lane&15.
// B 4x16 mirrors A under transpose, so B tiles load identically from E rows of bj.
__device__ __forceinline__ void ehtm_store_tile(const v8f& acc, int rbase, int col,
                                                const float* __restrict__ sq,
                                                float* __restrict__ dist) {
    float sqc = sq[col];
#pragma unroll
    for (int v = 0; v < 8; ++v) {
        int row = rbase + v;
        float d2 = sq[row] - 2.0f * acc[v] + sqc;
        float d = sqrtf(fmaxf(d2, 1e-8f));
        dist[(size_t)row * N_SZ + col] = d;
        dist[(size_t)col * N_SZ + row] = d;   // symmetry mirror
    }
}

__global__ __launch_bounds__(32) void ehtm_gemm_dist(const float* __restrict__ emb,
                                                     const float* __restrict__ sq,
                                                     float* __restrict__ dist) {
    int bi = blockIdx.y, bj = blockIdx.x;
    if (bj < bi) return;                      // upper triangle of 32x32 blocks
    int lane = threadIdx.x;
    int lm = lane & 15;
    int koff = (lane >> 4) << 1;              // 0 for lanes 0-15, 2 for lanes 16-31

    const float* pa0 = emb + (size_t)(bi * 32 + lm) * D_SZ + koff;
    const float* pa1 = emb + (size_t)(bi * 32 + 16 + lm) * D_SZ + koff;
    const float* pb0 = emb + (size_t)(bj * 32 + lm) * D_SZ + koff;
    const float* pb1 = emb + (size_t)(bj * 32 + 16 + lm) * D_SZ + koff;

    v8f acc00 = {0.f,0.f,0.f,0.f,0.f,0.f,0.f,0.f};
    v8f acc01 = acc00, acc10 = acc00, acc11 = acc00;

#pragma unroll 2
    for (int k = 0; k < D_SZ; k += 4) {
        v2f a0 = *(const v2f*)(pa0 + k);
        v2f a1 = *(const v2f*)(pa1 + k);
        v2f b0 = *(const v2f*)(pb0 + k);
        v2f b1 = *(const v2f*)(pb1 + k);
        acc00 = __builtin_amdgcn_wmma_f32_16x16x4_f32(false, a0, false, b0, (short)0, acc00, false, false);
        acc01 = __builtin_amdgcn_wmma_f32_16x16x4_f32(false, a0, false, b1, (short)0, acc01, false, false);
        acc10 = __builtin_amdgcn_wmma_f32_16x16x4_f32(false, a1, false, b0, (short)0, acc10, false, false);
        acc11 = __builtin_amdgcn_wmma_f32_16x16x4_f32(false, a1, false, b1, (short)0, acc11, false, false);
    }

    // C/D layout: lane<16 -> N=lane, M=v ; lane>=16 -> N=lane-16, M=v+8
    int rhalf = (lane >> 4) << 3;
    int r0 = bi * 32 + rhalf;                 // M-tile 0 row base
    int r1 = bi * 32 + 16 + rhalf;            // M-tile 1 row base
    int c0 = bj * 32 + lm;                    // N-tile 0 column
    int c1 = bj * 32 + 16 + lm;               // N-tile 1 column
    ehtm_store_tile(acc00, r0, c0, sq, dist);
    ehtm_store_tile(acc01, r0, c1, sq, dist);
    ehtm_store_tile(acc10, r1, c0, sq, dist);
    ehtm_store_tile(acc11, r1, c1, sq, dist);
}

// ---------------- batch-hard + semi-hard mining per anchor row ----------------
__global__ void ehtm_hard(const float* __restrict__ dist, const int* __restrict__ labels,
                          float* __restrict__ scal) {
    __shared__ float drow[N_SZ];
    __shared__ int   lab[N_SZ];
    __shared__ int   poslist[N_SZ];
    __shared__ float red[256];
    __shared__ int   npos;
    int i = blockIdx.x, t = threadIdx.x;

    for (int j = t; j < N_SZ; j += 256) {
        drow[j] = dist[(size_t)i * N_SZ + j];
        lab[j]  = labels[j];
    }
    if (t == 0) npos = 0;
    __syncthreads();

    int la = lab[i];
    float hp = -BIG_F, hn = BIG_F;
    for (int j = t; j < N_SZ; j += 256) {
        float d = drow[j];
        if (lab[j] == la) {
            if (j != i) {
                hp = fmaxf(hp, d);
                int k = atomicAdd(&npos, 1);
                poslist[k] = j;
            }
        } else {
            hn = fminf(hn, d);
        }
    }
    red[t] = hp; __syncthreads();
    for (int o = 128; o > 0; o >>= 1) { if (t < o) red[t] = fmaxf(red[t], red[t + o]); __syncthreads(); }
    hp = red[0]; __syncthreads();
    red[t] = hn; __syncthreads();
    for (int o = 128; o > 0; o >>= 1) { if (t < o) red[t] = fminf(red[t], red[t + o]); __syncthreads(); }
    hn = red[0]; __syncthreads();

    if (t == 0 && hp > -BIG_F && hn < BIG_F) {
        atomicAdd(&scal[1], fmaxf(hp - hn + MARGIN_F, 0.f));
        atomicAdd(&scal[2], 1.0f);
    }

    // semi-hard: for each positive j, smallest negative distance strictly > d(i,j)
    int P = npos;
    float shs = 0.f, shc = 0.f;
    for (int p = 0; p < P; ++p) {
        float q = drow[poslist[p]];
        float cand = BIG_F;
        for (int j = t; j < N_SZ; j += 256) {
            float d = drow[j];
            if (lab[j] != la && d > q) cand = fminf(cand, d);
        }
        red[t] = cand; __syncthreads();
        for (int o = 128; o > 0; o >>= 1) { if (t < o) red[t] = fminf(red[t], red[t + o]); __syncthreads(); }
        if (t == 0) {
            float cd = red[0];
            if (cd < BIG_F && cd < q + MARGIN_F) {
                shs += fmaxf(q - cd + MARGIN_F, 0.f);
                shc += 1.0f;
            }
        }
        __syncthreads();
    }
    if (t == 0 && shc > 0.f) {
        atomicAdd(&scal[3], shs);
        atomicAdd(&scal[4], shc);
    }
}

// ---------------- T = onehot^T * dist * onehot via LDS 64x64 accumulation ----------------
__global__ void ehtm_classT(const float* __restrict__ dist, const int* __restrict__ labels,
                            float* __restrict__ Tm) {
    __shared__ float Ts[NC * NC];
    int t = threadIdx.x;
    for (int k = t; k < NC * NC; k += 256) Ts[k] = 0.f;
    __syncthreads();
    size_t total = (size_t)N_SZ * N_SZ;
    for (size_t idx = (size_t)blockIdx.x * 256 + t; idx < total; idx += (size_t)gridDim.x * 256) {
        int i = (int)(idx / N_SZ);
        int j = (int)(idx - (size_t)i * N_SZ);
        atomicAdd(&Ts[labels[i] * NC + labels[j]], dist[idx]);
    }
    __syncthreads();
    for (int k = t; k < NC * NC; k += 256) {
        float v = Ts[k];
        if (v != 0.f) atomicAdd(&Tm[k], v);
    }
}

// ---------------- finalization ----------------
__global__ void ehtm_final(const float* __restrict__ Tm, const float* __restrict__ counts,
                           const int* __restrict__ in_anchor, const float* __restrict__ scal,
                           float* __restrict__ out) {
    __shared__ float red[NC];
    int c = threadIdx.x;
    float cnt = counts[c];
    int ia = in_anchor[c];
    float intra_c = 0.f;
    if (ia && cnt > 1.0f)
        intra_c = Tm[c * NC + c] / fmaxf(cnt * (cnt - 1.0f), 1.0f);
    float inter_c = 0.f;
    for (int c2 = c + 1; c2 < NC; ++c2) {
        float pn = cnt * counts[c2];
        if (ia && in_anchor[c2] && pn > 0.f) {
            float mi = Tm[c * NC + c2] / fmaxf(pn, 1.0f);
            inter_c += fmaxf(MARGIN_F - mi, 0.f);
        }
    }
    red[c] = 0.1f * intra_c + 0.1f * inter_c;   // L_INTRA = L_INTER = 0.1
    __syncthreads();
    if (c == 0) {
        float s = 0.f;
        for (int k = 0; k < NC; ++k) s += red[k];
        float trip = scal[0] / (float)B_SZ;
        float hm = (scal[2] > 0.f) ? scal[1] / fmaxf(scal[2], 1.f) : 0.f;
        float sm = (scal[4] > 0.f) ? scal[3] / fmaxf(scal[4], 1.f) : 0.f;
        out[0] = trip + hm + sm + s;
    }
}

extern "C" void kernel_launch(void* const* d_in, const int* in_sizes, int n_in,
                              void* d_out, int out_size, void* d_ws, size_t ws_size,
                              hipStream_t stream) {
    const float* anchor   = (const float*)d_in[0];
    const float* positive = (const float*)d_in[1];
    const float* negative = (const float*)d_in[2];
    const int*   al       = (const int*)d_in[3];
    const int*   nl       = (const int*)d_in[4];
    float* out = (float*)d_out;

    // workspace layout (floats): dist[N*N] | emb[N*D] | sq[N] | labels[N] |
    //                            T[64*64] | counts[64] | in_anchor[64] | scal[8]
    float* dist   = (float*)d_ws;
    float* emb    = dist + (size_t)N_SZ * N_SZ;
    float* sq     = emb + (size_t)N_SZ * D_SZ;
    int*   labels = (int*)(sq + N_SZ);
    float* Tm     = (float*)(labels + N_SZ);
    float* counts = Tm + NC * NC;
    int*   in_anc = (int*)(counts + NC);
    float* scal   = (float*)(in_anc + NC);

    int zero_n = NC * NC + NC + NC + 8;   // Tm..scal contiguous (int zeros == float zeros)
    ehtm_init<<<(zero_n + 255) / 256, 256, 0, stream>>>(Tm, zero_n);

    dim3 pg(D_SZ / 256, N_SZ);
    ehtm_pack<<<pg, 256, 0, stream>>>(anchor, positive, negative, emb);
    ehtm_labels<<<(N_SZ + 255) / 256, 256, 0, stream>>>(al, nl, labels, counts, in_anc);
    ehtm_norms<<<N_SZ, 256, 0, stream>>>(emb, sq, scal);

    dim3 gg(N_SZ / 32, N_SZ / 32);        // 96 x 96 blocks, upper triangle active
    ehtm_gemm_dist<<<gg, 32, 0, stream>>>(emb, sq, dist);

    ehtm_hard<<<B_SZ, 256, 0, stream>>>(dist, labels, scal);
    ehtm_classT<<<1024, 256, 0, stream>>>(dist, labels, Tm);
    ehtm_final<<<1, NC, 0, stream>>>(Tm, counts, in_anc, scal, out);
}